// KalmanFilter_72902774882340
// MI455X (gfx1250) — compile-verified
//
#include <hip/hip_runtime.h>

// Kalman filter scan on MI455X (gfx1250, wave32).
// One wave per batch element; covariance kept in WMMA C-layout registers,
// staged through per-block LDS for layout conversion between chained
// V_WMMA_F32_16X16X4_F32 products. All scalar dot-products run unguarded on
// all 32 lanes (clamped row indices); only final stores are lane-guarded, to
// avoid per-load exec-mask toggling on the serial critical path. The 8x8
// Gauss-Jordan pivot reciprocal uses v_rcp_f32 + one Newton step instead of
// the ~10-op IEEE divide sequence (pivots are well-conditioned: R >= 0.1*I).

typedef float v2f __attribute__((ext_vector_type(2)));
typedef float v8f __attribute__((ext_vector_type(8)));

#define BS 512
#define MM 8
#define TT 2048
#define DD 16

__device__ __forceinline__ v8f wmma4(v2f a, v2f b, v8f c) {
  // D = A(16x4 f32) * B(4x16 f32) + C(16x16 f32)
  return __builtin_amdgcn_wmma_f32_16x16x4_f32(
      /*neg_a=*/false, a, /*neg_b=*/false, b,
      /*c_mod=*/(short)0, c, /*reuse_a=*/false, /*reuse_b=*/false);
}

// Fast reciprocal: v_rcp_f32 (+1 Newton-Raphson refinement).
__device__ __forceinline__ float fastrcp(float x) {
  float y = __builtin_amdgcn_rcpf(x);
  return y * (2.0f - x * y);
}

// A-fragment of a 16x16 row-major matrix (optionally transposed).
// Lane: row r = lane&15 ; k = k0 + v + 2*(lane>=16), v in {0,1}.
__device__ __forceinline__ v2f fA(const float* Mtx, int k, int r, bool tr) {
  v2f a;
  if (!tr) { a.x = Mtx[r * 16 + k];  a.y = Mtx[r * 16 + k + 1]; }
  else     { a.x = Mtx[k * 16 + r];  a.y = Mtx[(k + 1) * 16 + r]; }
  return a;
}
// B-fragment of a 16x16 row-major matrix (optionally transposed).
// Lane: col c = lane&15 ; k = k0 + v + 2*(lane>=16).
__device__ __forceinline__ v2f fB(const float* Mtx, int k, int c, bool tr) {
  v2f b;
  if (!tr) { b.x = Mtx[k * 16 + c];       b.y = Mtx[(k + 1) * 16 + c]; }
  else     { b.x = Mtx[c * 16 + k];       b.y = Mtx[c * 16 + k + 1]; }
  return b;
}

// Full 16x16x16 f32 matmul = 4 chained WMMAs. Operand source variants:
__device__ __forceinline__ v8f mmLL(const float* A, bool tA, const float* B, bool tB,
                                    v8f c, int r, int h2) {
#pragma unroll
  for (int k0 = 0; k0 < 16; k0 += 4)
    c = wmma4(fA(A, k0 + h2, r, tA), fB(B, k0 + h2, r, tB), c);
  return c;
}
__device__ __forceinline__ v8f mmCL(const v2f aF[4], const float* B, bool tB,
                                    v8f c, int r, int h2) {
#pragma unroll
  for (int i = 0; i < 4; ++i)
    c = wmma4(aF[i], fB(B, 4 * i + h2, r, tB), c);
  return c;
}
__device__ __forceinline__ v8f mmLC(const float* A, bool tA, const v2f bF[4],
                                    v8f c, int r, int h2) {
#pragma unroll
  for (int i = 0; i < 4; ++i)
    c = wmma4(fA(A, 4 * i + h2, r, tA), bF[i], c);
  return c;
}
// Store a C-layout accumulator (16x16 f32) to LDS row-major.
__device__ __forceinline__ void storeC(float* Mtx, v8f c, int r, int hi8) {
#pragma unroll
  for (int v = 0; v < 8; ++v) Mtx[(v + hi8) * 16 + r] = c[v];
}

__global__ __launch_bounds__(32) void kf_kernel(
    const float* __restrict__ xg, const float* __restrict__ Fg,
    const float* __restrict__ Hg, const float* __restrict__ Qg,
    const float* __restrict__ Rg, float* __restrict__ outg) {
  __shared__ float sCov[256];   // canonical covariance (row-major 16x16)
  __shared__ float sTmp[256];   // staging for chained products
  __shared__ float sK[256];     // Kalman gain
  __shared__ float sA[256];     // A = I - K H
  __shared__ float sSinv[256];  // padded 16x16 S^-1 (only 8x8 block live)
  __shared__ float sMean[16];
  __shared__ float sMeanU[16];
  __shared__ float sResid[8];

  const int lane = threadIdx.x;
  const int b    = blockIdx.x;
  const int r    = lane & 15;          // row (A) or col (B/C) index
  const int r8   = lane & 7;           // clamped H-row index (all lanes safe)
  const int h2   = (lane >> 4) << 1;   // K sub-offset per lane half
  const int hi8  = (lane >> 4) << 3;   // C-layout row offset per lane half

  // ---- Constant fragments, loaded once from global into registers ----
  // Per-lane identity: A-frag(M) == B-frag(M^T). So HpA doubles as B-frag of
  // Hp^T, and FA doubles as B-frag of F^T.
  v2f HpA[4], HpB[4], FA[4], RpB[4];
  v8f RpC, QC, idC;
#pragma unroll
  for (int i = 0; i < 4; ++i) {
    int k = 4 * i + h2;
    HpA[i].x = (r < 8) ? Hg[r * 16 + k] : 0.f;       // Hp (rows 8..15 zero)
    HpA[i].y = (r < 8) ? Hg[r * 16 + k + 1] : 0.f;
    HpB[i].x = (k < 8) ? Hg[k * 16 + r] : 0.f;       // Hp as B operand
    HpB[i].y = (k + 1 < 8) ? Hg[(k + 1) * 16 + r] : 0.f;
    FA[i].x  = Fg[r * 16 + k];
    FA[i].y  = Fg[r * 16 + k + 1];
    RpB[i].x = (k < 8 && r < 8) ? Rg[k * 8 + r] : 0.f;   // padded R as B
    RpB[i].y = (k + 1 < 8 && r < 8) ? Rg[(k + 1) * 8 + r] : 0.f;
  }
#pragma unroll
  for (int v = 0; v < 8; ++v) {
    int row = v + hi8;
    RpC[v] = (row < 8 && r < 8) ? Rg[row * 8 + r] : 0.f;
    QC[v]  = Qg[row * 16 + r];
    idC[v] = (row == r) ? 1.f : 0.f;
  }

  // ---- Row caches for the scalar mat-vec work (all lanes, clamped rows) ----
  float hrow[16], frow[16];
#pragma unroll
  for (int j = 0; j < 16; ++j) {
    hrow[j] = Hg[r8 * 16 + j];   // row (lane&7) of H
    frow[j] = Fg[r * 16 + j];    // row (lane&15) of F
  }

  // ---- Initial state: mean0 = 0, cov0 = I ----
#pragma unroll
  for (int j = 0; j < 8; ++j) sSinv[lane * 8 + j] = 0.f;  // zero all 256
  if (lane < 16) sMean[lane] = 0.f;
  v8f covC = idC;
  storeC(sCov, covC, r, hi8);
  if (lane < 8) outg[(size_t)b * MM * TT + (size_t)lane * TT] = 0.f;
  __syncthreads();

  const v8f z = {0.f, 0.f, 0.f, 0.f, 0.f, 0.f, 0.f, 0.f};

  for (int t = 0; t < TT - 1; ++t) {
    // -- observation (all lanes load a valid clamped row), NaN mask --
    float o = xg[(size_t)b * MM * TT + (size_t)r8 * TT + t];
    const bool nl = (o != o);
    const int nanAny = __any((lane < 8) && nl);
    const float oc = nl ? 0.f : o;

    // -- residual: all lanes compute, lanes 0..7 store --
    float rs = oc;
#pragma unroll
    for (int j = 0; j < 16; ++j) rs -= hrow[j] * sMean[j];
    if (lane < 8) sResid[lane] = rs;

    // -- S = Hp cov Hp^T + Rp --
    v8f u = mmCL(HpA, sCov, false, z, r, h2);         // U = Hp * cov
    storeC(sTmp, u, r, hi8);
    __syncthreads();
    v8f S = mmLC(sTmp, false, HpA, RpC, r, h2);       // S = U * Hp^T + Rp
    storeC(sTmp, S, r, hi8);
    __syncthreads();

    // -- invert top-left 8x8 of S: lane-parallel Gauss-Jordan over 8 lanes --
    // (lanes 8..31 carry harmless duplicates; shuffles only source lanes 0..7)
    float sr[8], w[8];
#pragma unroll
    for (int j = 0; j < 8; ++j) {
      sr[j] = sTmp[r8 * 16 + j];
      w[j]  = (lane == j) ? 1.f : 0.f;
    }
#pragma unroll
    for (int k = 0; k < 8; ++k) {
      float pk  = __shfl(sr[k], k, 32);
      float inv = fastrcp(pk);
      float f   = sr[k] * inv;
#pragma unroll
      for (int j = 0; j < 8; ++j) {
        float skj = __shfl(sr[j], k, 32) * inv;
        float wkj = __shfl(w[j],  k, 32) * inv;
        bool me = (lane == k);
        sr[j] = me ? skj : (sr[j] - f * skj);
        w[j]  = me ? wkj : (w[j]  - f * wkj);
      }
    }
    if (lane < 8) {
#pragma unroll
      for (int j = 0; j < 8; ++j) sSinv[lane * 16 + j] = w[j];
    }
    __syncthreads();

    // -- K = cov Hp^T Sinv --
    v8f t1 = mmLC(sCov, false, HpA, z, r, h2);        // T1 = cov * Hp^T
    storeC(sTmp, t1, r, hi8);
    __syncthreads();
    v8f K = mmLL(sTmp, false, sSinv, false, z, r, h2);
    storeC(sK, K, r, hi8);
    __syncthreads();

    // -- A = I - K Hp --
    v8f kh = mmLC(sK, false, HpB, z, r, h2);          // K * Hp
    v8f am;
#pragma unroll
    for (int v = 0; v < 8; ++v) am[v] = idC[v] - kh[v];
    storeC(sA, am, r, hi8);
    __syncthreads();

    // -- cov_u = A cov A^T + K Rp K^T (Joseph form) --
    v8f t2 = mmLL(sA, false, sCov, false, z, r, h2);  // T2 = A * cov
    storeC(sTmp, t2, r, hi8);
    __syncthreads();
    v8f cu = mmLL(sTmp, false, sA, true, z, r, h2);   // T2 * A^T
    v8f t3 = mmLC(sK, false, RpB, z, r, h2);          // T3 = K * Rp
    storeC(sTmp, t3, r, hi8);
    __syncthreads();
    cu = mmLL(sTmp, false, sK, true, cu, r, h2);      // += T3 * K^T

    // -- NaN gate on covariance --
#pragma unroll
    for (int v = 0; v < 8; ++v) cu[v] = nanAny ? covC[v] : cu[v];

    // -- mean_u = mean + K resid: all lanes compute, lanes 0..15 store --
    {
      float acc = 0.f;
#pragma unroll
      for (int j = 0; j < 8; ++j) acc += sK[r * 16 + j] * sResid[j];
      float mu = nanAny ? sMean[r] : (sMean[r] + acc);
      if (lane < 16) sMeanU[lane] = mu;
    }
    storeC(sCov, cu, r, hi8);
    __syncthreads();

    // -- cov_p = F cov_u F^T + Q --
    v8f t4 = mmCL(FA, sCov, false, z, r, h2);         // T4 = F * cov_u
    storeC(sTmp, t4, r, hi8);
    __syncthreads();
    covC = mmLC(sTmp, false, FA, QC, r, h2);          // T4 * F^T + Q
    storeC(sCov, covC, r, hi8);

    // -- mean_p = F mean_u: all lanes compute, lanes 0..15 store --
    {
      float mp = 0.f;
#pragma unroll
      for (int j = 0; j < 16; ++j) mp += frow[j] * sMeanU[j];
      if (lane < 16) sMean[lane] = mp;
    }
    __syncthreads();

    // -- y[t+1] = H mean_p: all lanes compute, lanes 0..7 store --
    {
      float y = 0.f;
#pragma unroll
      for (int j = 0; j < 16; ++j) y += hrow[j] * sMean[j];
      if (lane < 8)
        outg[(size_t)b * MM * TT + (size_t)lane * TT + t + 1] = y;
    }
  }
}

extern "C" void kernel_launch(void* const* d_in, const int* in_sizes, int n_in,
                              void* d_out, int out_size, void* d_ws, size_t ws_size,
                              hipStream_t stream) {
  (void)in_sizes; (void)n_in; (void)out_size; (void)d_ws; (void)ws_size;
  const float* x = (const float*)d_in[0];
  const float* F = (const float*)d_in[1];
  const float* H = (const float*)d_in[2];
  const float* Q = (const float*)d_in[3];
  const float* R = (const float*)d_in[4];
  float* out = (float*)d_out;
  kf_kernel<<<dim3(BS), dim3(32), 0, stream>>>(x, F, H, Q, R, out);
}